// PostProcess_75170517614740
// MI455X (gfx1250) — compile-verified
//
#include <hip/hip_runtime.h>
#include <math.h>

// ---- problem constants (match reference) ----
#define BATCH 2
#define Hh 256
#define Ww 256
#define Gg 96
#define NC 12
#define KTOP 128
#define HW (Hh*Ww)               // 65536
#define G3 (Gg*Gg*Gg)            // 884736
#define LABEL_DIV 1000
#define STUFF_AREA 2048

typedef float v2f __attribute__((ext_vector_type(2)));
typedef float v8f __attribute__((ext_vector_type(8)));

// ---------------------------------------------------------------------------
// Kernel 0: zero the per-class counters
// ---------------------------------------------------------------------------
__global__ void k_init(int* counts2d, int* counts3d) {
  int i = threadIdx.x;
  if (i < BATCH * NC) { counts2d[i] = 0; counts3d[i] = 0; }
}

// ---------------------------------------------------------------------------
// Kernel 1: geometry truncation (occupancy<=0 -> TRUNCATION), streamed to out
// ---------------------------------------------------------------------------
__global__ void k_geom(const float* __restrict__ geom,
                       const float* __restrict__ occ,
                       float* __restrict__ out, int n) {
  int i = blockIdx.x * blockDim.x + threadIdx.x;
  if (i < n) out[i] = (occ[i] <= 0.0f) ? 3.0f : geom[i];
}

// ---------------------------------------------------------------------------
// Kernel 2: 2D prep — semantic argmax, class counts, 7x7 heatmap NMS
// ---------------------------------------------------------------------------
__global__ void k_prep2d(const float* __restrict__ sem_logits,
                         const float* __restrict__ heat,
                         float* __restrict__ hm_nms,
                         int* __restrict__ sem2d,
                         int* __restrict__ counts2d) {
  int i = blockIdx.x * blockDim.x + threadIdx.x;
  if (i >= BATCH * HW) return;
  int b = i / HW, p = i % HW, y = p / Ww, x = p % Ww;

  const float* sl = sem_logits + (size_t)b * NC * HW + p;
  float best = sl[0]; int bc = 0;
  #pragma unroll
  for (int c = 1; c < NC; c++) {
    float v = sl[(size_t)c * HW];
    if (v > best) { best = v; bc = c; }     // first-max tie break (argmax)
  }
  sem2d[i] = bc;
  atomicAdd(&counts2d[b * NC + bc], 1);

  const float* hb = heat + (size_t)b * HW;
  float h = hb[p];
  float mx = -INFINITY;
  for (int dy = -3; dy <= 3; dy++) {
    int yy = y + dy; if (yy < 0 || yy >= Hh) continue;
    for (int dx = -3; dx <= 3; dx++) {
      int xx = x + dx; if (xx < 0 || xx >= Ww) continue;
      float v = hb[yy * Ww + xx];
      mx = v > mx ? v : mx;
    }
  }
  hm_nms[i] = (mx == h) ? h : 0.0f;
}

// ---------------------------------------------------------------------------
// Kernel 3: top-K selection per batch (iterative argmax == lax.top_k order)
// ---------------------------------------------------------------------------
__global__ void k_topk(float* __restrict__ hm_nms,
                       const int* __restrict__ sem2d,
                       float* __restrict__ cy, float* __restrict__ cx,
                       int* __restrict__ ccls, int* __restrict__ cval,
                       int* __restrict__ anyv,
                       float* __restrict__ out_centers,   // [B,K,2]
                       float* __restrict__ out_classes,   // [B,K]
                       float* __restrict__ out_scores) {  // [B,K]
  int b = blockIdx.x;
  __shared__ float sv[256];
  __shared__ int   si[256];
  float* nms = hm_nms + (size_t)b * HW;
  const int* sem = sem2d + (size_t)b * HW;
  int anyl = 0;

  for (int k = 0; k < KTOP; k++) {
    float best = -INFINITY; int bidx = 0x7fffffff;
    for (int i = threadIdx.x; i < HW; i += blockDim.x) {
      float v = nms[i];
      if (v > best) { best = v; bidx = i; }
    }
    sv[threadIdx.x] = best; si[threadIdx.x] = bidx;
    __syncthreads();
    if (threadIdx.x == 0) {
      float gb = sv[0]; int gi = si[0];
      for (int t = 1; t < 256; t++)
        if (sv[t] > gb || (sv[t] == gb && si[t] < gi)) { gb = sv[t]; gi = si[t]; }
      nms[gi] = -INFINITY;                 // mask for next iteration
      float fy = (float)(gi / Ww), fx = (float)(gi % Ww);
      int cls = sem[gi];
      int vld = (gb > 0.1f) ? 1 : 0;
      anyl |= vld;
      cy[b * KTOP + k] = fy;  cx[b * KTOP + k] = fx;
      ccls[b * KTOP + k] = cls; cval[b * KTOP + k] = vld;
      out_centers[((size_t)b * KTOP + k) * 2 + 0] = fy;
      out_centers[((size_t)b * KTOP + k) * 2 + 1] = fx;
      out_classes[b * KTOP + k] = (float)cls;
      out_scores[b * KTOP + k] = gb;
    }
    __syncthreads();
  }
  if (threadIdx.x == 0) anyv[b] = anyl;
}

// ---------------------------------------------------------------------------
// WMMA distance argmin core: one wave handles 16 points x 128 centers using
// 8x V_WMMA_F32_16X16X4_F32.  d2 = [py px py^2+px^2 1] . [-2cy -2cx 1 cy^2+cx^2]
// A 16x4 layout: lanes 0-15 hold K=0,1 ; lanes 16-31 hold K=2,3.
// D 16x16 layout: lane holds column N=lane%16; d[r] is row r (+8 for lanes>=16).
// B fragments for all 8 center tiles are built branchlessly (no EXEC churn)
// and hoisted out of the WMMA loop so the 8 WMMAs issue back-to-back with only
// independent VALU min-updates interleaved (co-execution friendly).
// ---------------------------------------------------------------------------
__device__ __forceinline__ void wmma_argmin128(
    float py, float px, int lane,
    const float* __restrict__ bcy, const float* __restrict__ bcx,
    const int* __restrict__ bval,
    float bestV[8], int bestI[8]) {
  const int m = lane & 15;
  const bool lo = (lane < 16);

  v2f a;
  a.x = lo ? py : fmaf(py, py, px * px);
  a.y = lo ? px : 1.0f;

  // Build all 8 B fragments up front; every lane loads (no divergent loads).
  v2f bfrag[8];
  #pragma unroll
  for (int t = 0; t < 8; t++) {
    int c = t * 16 + m;
    float ccy = bcy[c];
    float ccx = bcx[c];
    int   vld = bval[c];
    float norm = fmaf(ccy, ccy, ccx * ccx);
    float masked = vld ? norm : 3.0e30f;     // jnp.where(valid, d2, inf)
    bfrag[t].x = lo ? (-2.0f * ccy) : 1.0f;
    bfrag[t].y = lo ? (-2.0f * ccx) : masked;
  }

  #pragma unroll
  for (int r = 0; r < 8; r++) { bestV[r] = INFINITY; bestI[r] = 0; }

  #pragma unroll
  for (int t = 0; t < 8; t++) {
    v8f d = {};
    d = __builtin_amdgcn_wmma_f32_16x16x4_f32(
        /*neg_a=*/false, a, /*neg_b=*/false, bfrag[t],
        /*c_mod=*/(short)0, d, /*reuse_a=*/false, /*reuse_b=*/false);
    int idx = t * 16 + m;
    #pragma unroll
    for (int r = 0; r < 8; r++) {
      float v = d[r];
      if (v < bestV[r]) { bestV[r] = v; bestI[r] = idx; }  // t ascending -> first-index ties
    }
  }

  // min-reduce across the 16 lanes of each half (xor<=8 stays in-half, wave32)
  #pragma unroll
  for (int s = 1; s < 16; s <<= 1) {
    #pragma unroll
    for (int r = 0; r < 8; r++) {
      float ov = __shfl_xor(bestV[r], s, 32);
      int   oi = __shfl_xor(bestI[r], s, 32);
      if (ov < bestV[r] || (ov == bestV[r] && oi < bestI[r])) {
        bestV[r] = ov; bestI[r] = oi;
      }
    }
  }
}

// ---------------------------------------------------------------------------
// Kernel 4: 2D instance assignment + panoptic fusion  (wave per 16-pixel tile)
// ---------------------------------------------------------------------------
__global__ void k_assign2d(const float* __restrict__ offset2d,
                           const int* __restrict__ sem2d,
                           const int* __restrict__ counts2d,
                           const float* __restrict__ cy, const float* __restrict__ cx,
                           const int* __restrict__ ccls, const int* __restrict__ cval,
                           const int* __restrict__ anyv,
                           float* __restrict__ pan2d_out) {
  int lane = threadIdx.x & 31;
  int wave = threadIdx.x >> 5;
  int tile = blockIdx.x * (blockDim.x >> 5) + wave;   // exact grid, no bounds check
  int b  = tile / (HW / 16);
  int p0 = (tile % (HW / 16)) * 16;
  int m  = lane & 15;
  int p  = p0 + m;
  int y  = p / Ww, x = p % Ww;
  const float* off = offset2d + (size_t)b * 2 * HW;
  float py = (float)y + off[p];
  float px = (float)x + off[HW + p];

  float bestV[8]; int bestI[8];
  wmma_argmin128(py, px, lane, cy + b * KTOP, cx + b * KTOP, cval + b * KTOP,
                 bestV, bestI);

  int half = lane >> 4;
  int rl = lane & 15;
  if (rl < 8) {
    int row = half * 8 + rl;
    int inst = bestI[rl];
    int pw = p0 + row;
    int sem = sem2d[(size_t)b * HW + pw];
    bool thing = (sem >= 1 && sem <= 7);
    int pan;
    if (thing) {
      pan = anyv[b] ? (ccls[b * KTOP + inst] * LABEL_DIV + inst + 1) : 0;
    } else {
      pan = (counts2d[b * NC + sem] >= STUFF_AREA) ? sem * LABEL_DIV : 0;
    }
    pan2d_out[(size_t)b * HW + pw] = (float)pan;
  }
}

// ---------------------------------------------------------------------------
// Kernel 5: 3D prep — semantic argmax + fg-weighted class counts
// ---------------------------------------------------------------------------
__global__ void k_prep3d(const float* __restrict__ sem_logits,
                         const float* __restrict__ geom,
                         const float* __restrict__ occ,
                         int* __restrict__ sem3d,
                         int* __restrict__ counts3d) {
  int i = blockIdx.x * blockDim.x + threadIdx.x;
  if (i >= BATCH * G3) return;
  int b = i / G3; int v = i % G3;
  const float* sl = sem_logits + (size_t)b * NC * G3 + v;
  float best = sl[0]; int bc = 0;
  #pragma unroll
  for (int c = 1; c < NC; c++) {
    float t = sl[(size_t)c * G3];
    if (t > best) { best = t; bc = c; }
  }
  sem3d[i] = bc;
  float g = (occ[i] <= 0.0f) ? 3.0f : geom[i];
  if (fabsf(g) < 1.5f) atomicAdd(&counts3d[b * NC + bc], 1);
}

// ---------------------------------------------------------------------------
// Kernel 6: 3D voxel projection + assignment + panoptic fusion
// ---------------------------------------------------------------------------
__global__ void k_assign3d(const float* __restrict__ offset3d,
                           const float* __restrict__ intrinsic,
                           const float* __restrict__ geom,
                           const float* __restrict__ occ,
                           const int* __restrict__ sem3d,
                           const int* __restrict__ counts3d,
                           const float* __restrict__ cy, const float* __restrict__ cx,
                           const int* __restrict__ ccls, const int* __restrict__ cval,
                           const int* __restrict__ anyv,
                           float* __restrict__ pan3d_out) {
  int lane = threadIdx.x & 31;
  int wave = threadIdx.x >> 5;
  int tile = blockIdx.x * (blockDim.x >> 5) + wave;   // exact grid
  int b  = tile / (G3 / 16);
  int v0 = (tile % (G3 / 16)) * 16;
  int m  = lane & 15;
  int vv = v0 + m;
  int i3 = vv / (Gg * Gg), j3 = (vv / Gg) % Gg, k3 = vv % Gg;
  const float* off = offset3d + (size_t)b * 3 * G3;
  float o0 = off[vv], o1 = off[G3 + vv], o2 = off[2 * (size_t)G3 + vv];
  const float* intr = intrinsic + (size_t)b * 9;
  float fx = intr[0], u0c = intr[2], fy = intr[4], v0c = intr[5];
  const float half_m = Gg * 0.0625f * 0.5f;           // 3.0 m
  float Xm = ((float)i3 + o0 + 0.5f) * 0.0625f - half_m;
  float Ym = ((float)j3 + o1 + 0.5f) * 0.0625f - half_m;
  float Zm = ((float)k3 + o2 + 0.5f) * 0.0625f;       // DEPTH_MIN = 0
  Zm = fminf(fmaxf(Zm, 0.1f), 6.0f);
  float u = fx * Xm / Zm + u0c;      // "px" vs cx
  float v = fy * Ym / Zm + v0c;      // "py" vs cy

  float bestV[8]; int bestI[8];
  wmma_argmin128(v, u, lane, cy + b * KTOP, cx + b * KTOP, cval + b * KTOP,
                 bestV, bestI);

  int half = lane >> 4;
  int rl = lane & 15;
  if (rl < 8) {
    int row = half * 8 + rl;
    int inst = bestI[rl];
    int vw = v0 + row;
    size_t gi = (size_t)b * G3 + vw;
    int sem = sem3d[gi];
    float g = (occ[gi] <= 0.0f) ? 3.0f : geom[gi];
    bool fg = fabsf(g) < 1.5f;
    bool thing = (sem >= 1 && sem <= 7);
    int pan;
    if (thing) {
      pan = anyv[b] ? (ccls[b * KTOP + inst] * LABEL_DIV + inst + 1) : 0;
    } else {
      pan = (counts3d[b * NC + sem] >= STUFF_AREA) ? sem * LABEL_DIV : 0;
    }
    pan3d_out[gi] = fg ? (float)pan : 0.0f;
  }
}

// ---------------------------------------------------------------------------
// Host-side launch
// ---------------------------------------------------------------------------
extern "C" void kernel_launch(void* const* d_in, const int* in_sizes, int n_in,
                              void* d_out, int out_size, void* d_ws, size_t ws_size,
                              hipStream_t stream) {
  const float* semantic2d = (const float*)d_in[0];
  const float* center2d   = (const float*)d_in[1];
  const float* offset2d   = (const float*)d_in[2];
  const float* geometry   = (const float*)d_in[3];
  const float* occupancy  = (const float*)d_in[4];
  const float* semantic3d = (const float*)d_in[5];
  const float* offset3d   = (const float*)d_in[6];
  const float* intrinsic  = (const float*)d_in[7];

  float* out = (float*)d_out;
  // output layout (floats), concatenated in reference return order
  float* o_pan2d   = out;                         // B*H*W      = 131072
  float* o_pan3d   = out + 131072;                // B*G^3      = 1769472
  float* o_geom    = out + 1900544;               // B*G^3      = 1769472
  float* o_centers = out + 3670016;               // B*K*2      = 512
  float* o_classes = out + 3670528;               // B*K        = 256
  float* o_scores  = out + 3670784;               // B*K        = 256

  // workspace layout (4-byte elements)
  float* ws_f = (float*)d_ws;
  int*   ws_i = (int*)d_ws;
  float* hm_nms   = ws_f + 0;        // 131072
  int*   sem2d    = ws_i + 131072;   // 131072
  int*   counts2d = ws_i + 262144;   // 24
  int*   counts3d = ws_i + 262176;   // 24
  float* w_cy     = ws_f + 262208;   // 256
  float* w_cx     = ws_f + 262464;   // 256
  int*   w_cls    = ws_i + 262720;   // 256
  int*   w_val    = ws_i + 262976;   // 256
  int*   w_any    = ws_i + 263232;   // 2
  int*   sem3d    = ws_i + 263296;   // 1769472  (total ~8.1 MB)

  k_init<<<1, 64, 0, stream>>>(counts2d, counts3d);

  k_geom<<<(BATCH * G3 + 255) / 256, 256, 0, stream>>>(
      geometry, occupancy, o_geom, BATCH * G3);

  k_prep2d<<<(BATCH * HW + 255) / 256, 256, 0, stream>>>(
      semantic2d, center2d, hm_nms, sem2d, counts2d);

  k_topk<<<BATCH, 256, 0, stream>>>(
      hm_nms, sem2d, w_cy, w_cx, w_cls, w_val, w_any,
      o_centers, o_classes, o_scores);

  // 2D assign: 8192 tiles of 16 px, 8 waves/block -> 1024 blocks (exact)
  k_assign2d<<<(BATCH * HW / 16) / 8, 256, 0, stream>>>(
      offset2d, sem2d, counts2d, w_cy, w_cx, w_cls, w_val, w_any, o_pan2d);

  k_prep3d<<<(BATCH * G3 + 255) / 256, 256, 0, stream>>>(
      semantic3d, geometry, occupancy, sem3d, counts3d);

  // 3D assign: 110592 tiles, 8 waves/block -> 13824 blocks (exact)
  k_assign3d<<<(BATCH * G3 / 16) / 8, 256, 0, stream>>>(
      offset3d, intrinsic, geometry, occupancy, sem3d, counts3d,
      w_cy, w_cx, w_cls, w_val, w_any, o_pan3d);
}